// RuleBasedLoss_36266703847674
// MI455X (gfx1250) — compile-verified
//
#include <hip/hip_runtime.h>
#include <math.h>

// Problem constants (match the reference)
#define LENGTH 48
#define BB 16
#define KK 4
#define NMAX 47                       // LENGTH-1
#define SS (LENGTH * (LENGTH - 1) / 2) // 1128 spans
#define SB (SS * BB)                   // 18048 (s,b) problems
#define WAVES_PER_BLOCK 8
#define BLOCK_THREADS (WAVES_PER_BLOCK * 32)

typedef __attribute__((ext_vector_type(2))) float v2f;
typedef __attribute__((ext_vector_type(8))) float v8f;

__device__ __forceinline__ float wave_sum(float v) {
#pragma unroll
  for (int off = 16; off > 0; off >>= 1) v += __shfl_xor(v, off, 32);
  return v;
}
__device__ __forceinline__ float wave_max(float v) {
#pragma unroll
  for (int off = 16; off > 0; off >>= 1) v = fmaxf(v, __shfl_xor(v, off, 32));
  return v;
}
// Gather value for column idx (0..46): column n lives on lane n%32, slot n/32.
__device__ __forceinline__ float wave_get(float v0, float v1, int idx) {
  float a = __shfl(v0, idx & 31, 32);
  float b = __shfl(v1, idx & 31, 32);
  return (idx < 32) ? a : b;
}

// One wave32 per (s,b). Writes per-problem partial loss (unscaled) to partial[flat].
__global__ __launch_bounds__(BLOCK_THREADS) void rb_span_kernel(
    const float* __restrict__ scalars, const int* __restrict__ rules,
    const float* __restrict__ mask, const int* __restrict__ ridx,
    const int* __restrict__ levels, const float* __restrict__ emb,
    float* __restrict__ partial) {
  __shared__ float stage[WAVES_PER_BLOCK][64];
  const int wave = threadIdx.x >> 5;
  const int lane = threadIdx.x & 31;
  const int flat = blockIdx.x * WAVES_PER_BLOCK + wave;  // = s*B + b (row-major)
  if (flat >= SB) return;  // uniform per wave
  const int s = flat / BB;
  const int level = levels[s];  // 1..47 valid split columns
  const float* row = scalars + (size_t)flat * NMAX;
  float* lds = &stage[wave][0];

  // --- Stage the 47-float scalar row into LDS via CDNA5 async global->LDS ---
#if __has_builtin(__builtin_amdgcn_global_load_async_to_lds_b32)
  {
    // clang declares the builtin as (as1 int*, as3 int*, imm offset, imm cpol)
    typedef __attribute__((address_space(1))) int GI;
    typedef __attribute__((address_space(3))) int LI;
    __builtin_amdgcn_global_load_async_to_lds_b32((GI*)(row + lane),
                                                  (LI*)(lds + lane), 0, 0);
    if (lane < NMAX - 32)
      __builtin_amdgcn_global_load_async_to_lds_b32((GI*)(row + 32 + lane),
                                                    (LI*)(lds + 32 + lane), 0, 0);
  }
#if __has_builtin(__builtin_amdgcn_s_wait_asynccnt)
  __builtin_amdgcn_s_wait_asynccnt(0);
#else
  asm volatile("s_wait_asynccnt 0" ::: "memory");
#endif
  asm volatile("" ::: "memory");  // compiler barrier: LDS now holds the row
#else
  lds[lane] = row[lane];
  if (lane < NMAX - 32) lds[32 + lane] = row[32 + lane];
#endif

  const bool v0 = lane < level;
  const bool v1 = (lane + 32) < level;
  const float x0 = lds[lane];
  const float x1 = (lane < NMAX - 32) ? lds[32 + lane] : 0.0f;

  // softmax over valid columns (padded columns -> exactly 0, as reference)
  const float m = wave_max(fmaxf(v0 ? x0 : -3.0e38f, v1 ? x1 : -3.0e38f));
  const float e0 = v0 ? __expf(x0 - m) : 0.0f;
  const float e1 = v1 ? __expf(x1 - m) : 0.0f;
  const float inv_se = 1.0f / wave_sum(e0 + e1);
  const float sp0 = e0 * inv_se;
  const float sp1 = e1 * inv_se;

  // K=4 rule entries (wave-uniform scalar data)
  const int base = flat * KK;
  int id[KK];
  float mk[KK], mr[KK];
#pragma unroll
  for (int k = 0; k < KK; ++k) {
    id[k] = ridx[base + k];          // guaranteed < level
    mk[k] = mask[base + k];          // 0.0 or 1.0
    mr[k] = emb[rules[base + k]] * mk[k];
  }

  // rule_lab scatter onto this lane's two columns; exp-normalize over valid cols
  float rl0 = 0.f, rl1 = 0.f;
#pragma unroll
  for (int k = 0; k < KK; ++k) {
    rl0 += (id[k] == lane) ? mr[k] : 0.f;
    rl1 += (id[k] == lane + 32) ? mr[k] : 0.f;
  }
  const float er0 = v0 ? __expf(fabsf(rl0)) : 0.f;  // non-scattered valid -> 1
  const float er1 = v1 ? __expf(fabsf(rl1)) : 0.f;
  const float invE = 1.0f / (wave_sum(er0 + er1) + 1e-10f);

  // values at the <=4 scattered indices (diff_mask support)
  float rlv[KK], spv[KK], dmv[KK];
  bool first[KK];
#pragma unroll
  for (int k = 0; k < KK; ++k) {
    float rsum = 0.f, dsum = 0.f;
#pragma unroll
    for (int k2 = 0; k2 < KK; ++k2)
      if (id[k2] == id[k]) { rsum += mr[k2]; dsum += mk[k2]; }
    rlv[k] = __expf(fabsf(rsum)) * invE;
    dmv[k] = dsum;
    spv[k] = wave_get(sp0, sp1, id[k]);
    bool f = true;
    for (int k2 = 0; k2 < k; ++k2)
      if (id[k2] == id[k]) f = false;
    first[k] = f;  // dedupe: count each matrix position (i,j) once
  }

  // <=6 off-diagonal index pairs; diagonal contributes exactly 0
  float acc = 0.f;
#pragma unroll
  for (int ka = 0; ka < KK; ++ka)
#pragma unroll
    for (int kb = ka + 1; kb < KK; ++kb) {
      if (!first[ka] || !first[kb]) continue;
      if (id[ka] == id[kb]) continue;
      if (!(dmv[ka] > 0.f && dmv[kb] > 0.f)) continue;  // sign mask
      const int lo = (id[ka] < id[kb]) ? ka : kb;
      const int hi = (id[ka] < id[kb]) ? kb : ka;
      const float trl = tanhf(50.0f * (rlv[hi] - rlv[lo]));
      const float tsp = tanhf(50.0f * (spv[hi] - spv[lo]));
      const float d = trl - tsp;
      acc += d * d;
    }
  if (lane == 0) partial[flat] = acc;
}

// Deterministic reduction of SB partials -> out[0] = 0.01 * sum.
// 256-thread fixed-order strided sum, then a WMMA f32 16x16x4 row-sum engine
// (B = ones) collapses the 256 block partials; D column 0 is combined from
// lanes 0 (M=0..7) and 16 (M=8..15) per the C/D VGPR layout.
__global__ __launch_bounds__(256) void rb_reduce_kernel(
    const float* __restrict__ partial, float* __restrict__ out) {
  __shared__ float sm[256];
  float a = 0.f;
  for (int i = threadIdx.x; i < SB; i += 256) a += partial[i];
  sm[threadIdx.x] = a;
  __syncthreads();
#if __has_builtin(__builtin_amdgcn_wmma_f32_16x16x4_f32)
  if (threadIdx.x < 32) {  // full wave -> EXEC all ones for WMMA
    const int lane = threadIdx.x;
    const int mrow = lane & 15;
    const int kh = (lane >> 4) << 1;  // lanes 0-15: K=0,1 ; lanes 16-31: K=2,3
    v8f acc = {0.f, 0.f, 0.f, 0.f, 0.f, 0.f, 0.f, 0.f};
    v2f ones = {1.0f, 1.0f};
#pragma unroll
    for (int c = 0; c < 4; ++c) {  // four 16x4 chunks of the 256 partials
      v2f am;
      am.x = sm[c * 64 + (kh + 0) * 16 + mrow];
      am.y = sm[c * 64 + (kh + 1) * 16 + mrow];
      acc = __builtin_amdgcn_wmma_f32_16x16x4_f32(false, am, false, ones,
                                                  (short)0, acc, false, false);
    }
    float ssum = acc[0] + acc[1] + acc[2] + acc[3] +
                 acc[4] + acc[5] + acc[6] + acc[7];
    ssum += __shfl(ssum, 16, 32);  // rows 0-7 (lane 0) + rows 8-15 (lane 16)
    if (lane == 0) out[0] = 0.01f * ssum;
  }
#else
  for (int stride = 128; stride > 0; stride >>= 1) {
    if ((int)threadIdx.x < stride) sm[threadIdx.x] += sm[threadIdx.x + stride];
    __syncthreads();
  }
  if (threadIdx.x == 0) out[0] = 0.01f * sm[0];
#endif
}

extern "C" void kernel_launch(void* const* d_in, const int* in_sizes, int n_in,
                              void* d_out, int out_size, void* d_ws,
                              size_t ws_size, hipStream_t stream) {
  (void)in_sizes; (void)n_in; (void)out_size; (void)ws_size;
  const float* scalars = (const float*)d_in[0];   // [S,B,NMAX] f32
  const int* rules     = (const int*)d_in[1];     // [S,B,K]
  const float* maskp   = (const float*)d_in[2];   // [S,B,K] f32
  const int* ridx      = (const int*)d_in[3];     // [S,B,K]
  const int* levels    = (const int*)d_in[4];     // [S]
  const float* emb     = (const float*)d_in[5];   // [VOCAB,1] f32
  float* out = (float*)d_out;
  float* partial = (float*)d_ws;  // needs SB*4 = 72,192 bytes of scratch

  dim3 grid((SB + WAVES_PER_BLOCK - 1) / WAVES_PER_BLOCK);
  rb_span_kernel<<<grid, BLOCK_THREADS, 0, stream>>>(scalars, rules, maskp,
                                                     ridx, levels, emb, partial);
  rb_reduce_kernel<<<1, 256, 0, stream>>>(partial, out);
}